// temporalAttention_89644557403042
// MI455X (gfx1250) — compile-verified
//
#include <hip/hip_runtime.h>

#define B_ 64
#define F_ 128
#define T_ 64
#define D_ 256

typedef __attribute__((ext_vector_type(2))) float v2f;
typedef __attribute__((ext_vector_type(8))) float v8f;

__device__ __forceinline__ float sigmoidf(float v) {
    return 1.0f / (1.0f + __expf(-v));
}

// ---------------------------------------------------------------------------
// Kernel 1: per-timestep contraction scalars
//   a_t = wconv[t].w1[t], c_t = bconv[t].w1[t]
//   u_t = wconv[t].w2,    v_t = bconv[t].w2
// ws layout: [a(64) | c(64) | u(64) | v(64) | P(64*128)]
// ---------------------------------------------------------------------------
__global__ void k_scalars(const float* __restrict__ wconv,
                          const float* __restrict__ bconv,
                          const float* __restrict__ w1,
                          const float* __restrict__ w2,
                          float* __restrict__ ws) {
    int t = threadIdx.x;
    if (t >= T_) return;
    const float* wc  = wconv + t * D_;
    const float* bc  = bconv + t * D_;
    const float* w1t = w1    + t * D_;
    float a = 0.f, c = 0.f, u = 0.f, v = 0.f;
    for (int d = 0; d < D_; ++d) {
        float wcd = wc[d], bcd = bc[d], w1d = w1t[d], w2d = w2[d];
        a = fmaf(wcd, w1d, a);
        c = fmaf(bcd, w1d, c);
        u = fmaf(wcd, w2d, u);
        v = fmaf(bcd, w2d, v);
    }
    ws[t] = a; ws[64 + t] = c; ws[128 + t] = u; ws[192 + t] = v;
}

// ---------------------------------------------------------------------------
// Kernel 2: per (b,t): vatt over f, S0/S1 reduction, tatt, scaled attention
// out, and P[b, 0:64]=tatt*S1, P[b, 64:128]=tatt*S0.
// One block per b; threadIdx.x = t (coalesced along innermost t dim).
// vatt cached in LDS (32 KB) to avoid recomputing sigmoid for the scale pass.
// ---------------------------------------------------------------------------
__global__ void k_atten(const float* __restrict__ x,
                        const float* __restrict__ b1,
                        const float* __restrict__ b2,
                        const float* __restrict__ ws,
                        float* __restrict__ P,
                        float* __restrict__ attn) {
    __shared__ float vatt_s[F_ * T_];   // [f][t], t contiguous -> bank friendly
    int b = blockIdx.x;
    int t = threadIdx.x;                // 0..63 (2 waves)
    const float* xb = x    + (size_t)b * F_ * T_;
    float*       ab = attn + (size_t)b * F_ * T_;

    float a_t = ws[t], c_t = ws[64 + t], u_t = ws[128 + t], v_t = ws[192 + t];

    float s0 = 0.f, s1 = 0.f;
    for (int f = 0; f < F_; ++f) {
        float xv  = xb[f * T_ + t];                       // coalesced over t
        float arg = fmaf(xv, a_t, c_t) + b1[t * F_ + f];
        float vv  = sigmoidf(arg);
        vatt_s[f * T_ + t] = vv;
        s0 += vv;
        s1  = fmaf(vv, xv, s1);
    }

    float tatt = sigmoidf(fmaf(s1, u_t, fmaf(s0, v_t, b2[t])));

    for (int f = 0; f < F_; ++f) {
        ab[f * T_ + t] = vatt_s[f * T_ + t] * tatt;       // coalesced over t
    }

    P[b * (2 * T_) + t]      = tatt * s1;
    P[b * (2 * T_) + T_ + t] = tatt * s0;
}

// ---------------------------------------------------------------------------
// Kernel 3: out[64,256] = P[64,128] @ Wcat[128,256] with f32 WMMA 16x16x4.
// Wcat row k = wconv[k] for k<64, bconv[k-64] otherwise.
// One wave (32 threads) per 16x16 output tile; grid = (4, 16) tiles.
// A frag (16x4): lanes 0-15 hold K=k,k+1; lanes 16-31 hold K=k+2,k+3.
// B frag (4x16): same K split, N on lanes.
// C/D (16x16 f32): VGPR r, lanes 0-15 -> M=r, lanes 16-31 -> M=8+r.
// ---------------------------------------------------------------------------
__global__ void k_out_wmma(const float* __restrict__ P,
                           const float* __restrict__ wconv,
                           const float* __restrict__ bconv,
                           float* __restrict__ out) {
    int lane  = threadIdx.x;        // 0..31
    int tileM = blockIdx.x;         // 0..3
    int tileN = blockIdx.y;         // 0..15
    int half  = lane >> 4;          // K sub-pair select
    int l16   = lane & 15;

    int rowA = tileM * 16 + l16;    // M index this lane supplies for A
    int col  = tileN * 16 + l16;    // N index this lane supplies for B / D

    v8f acc = {};
    for (int k = 0; k < 2 * T_; k += 4) {
        int ka = k + 2 * half;

        v2f afrag;
        afrag.x = P[rowA * (2 * T_) + ka];
        afrag.y = P[rowA * (2 * T_) + ka + 1];

        const float* r0 = (ka     < T_) ? (wconv + ka * D_)
                                        : (bconv + (ka - T_) * D_);
        const float* r1 = (ka + 1 < T_) ? (wconv + (ka + 1) * D_)
                                        : (bconv + (ka + 1 - T_) * D_);
        v2f bfrag;
        bfrag.x = r0[col];
        bfrag.y = r1[col];

        acc = __builtin_amdgcn_wmma_f32_16x16x4_f32(
            /*neg_a=*/false, afrag, /*neg_b=*/false, bfrag,
            /*c_mod=*/(short)0, acc, /*reuse_a=*/false, /*reuse_b=*/false);
    }

#pragma unroll
    for (int r = 0; r < 8; ++r) {
        int row = tileM * 16 + r + 8 * half;
        out[row * D_ + col] = acc[r];
    }
}

// ---------------------------------------------------------------------------
extern "C" void kernel_launch(void* const* d_in, const int* in_sizes, int n_in,
                              void* d_out, int out_size, void* d_ws, size_t ws_size,
                              hipStream_t stream) {
    const float* x     = (const float*)d_in[0];   // [B,F,T]
    const float* wconv = (const float*)d_in[1];   // [T,D]
    const float* bconv = (const float*)d_in[2];   // [T,D]
    const float* w1    = (const float*)d_in[3];   // [T,D]
    const float* b1    = (const float*)d_in[4];   // [T,F]
    const float* w2    = (const float*)d_in[5];   // [D]
    const float* b2    = (const float*)d_in[6];   // [T]

    float* out  = (float*)d_out;          // out[B,D] = 16384 floats
    float* attn = out + B_ * D_;          // variAtten*tatt [B,F,T]

    float* ws = (float*)d_ws;             // scalars (256) + P (64*128)
    float* P  = ws + 256;

    k_scalars<<<1, 64, 0, stream>>>(wconv, bconv, w1, w2, ws);
    k_atten<<<B_, T_, 0, stream>>>(x, b1, b2, ws, P, attn);
    k_out_wmma<<<dim3(B_ / 16, D_ / 16), 32, 0, stream>>>(P, wconv, bconv, out);
}